// GridEncoding3D_84670985273547
// MI455X (gfx1250) — compile-verified
//
#include <hip/hip_runtime.h>

#define NBINS 128
#define NB1   (NBINS - 1)

// Native clang vector type: accepted by __builtin_nontemporal_* and lowers to
// single b128 global loads/stores.
typedef float v4f __attribute__((ext_vector_type(4)));

// ---------------------------------------------------------------------------
// GridEncoding3D: trilinear interpolation of a [128,128,128,4] fp32 grid at
// N fp32 points in [0,1)^3.  Memory-bound gather kernel: grid (32 MB) is
// L2-resident on MI455X (192 MB L2); HBM floor ~88 MB total traffic.
// One point per thread, 8x global_load_b128 gathers, NT streaming store.
// ---------------------------------------------------------------------------

__global__ __launch_bounds__(256) void grid_enc3d_trilerp(
    const float* __restrict__ x,      // [N,3]
    const float* __restrict__ grid,   // [128,128,128,4] -> v4f[128*128*128]
    float* __restrict__ out,          // [N,4]
    int n)
{
    const v4f* __restrict__ g4 = reinterpret_cast<const v4f*>(grid);
    v4f* __restrict__ o4       = reinterpret_cast<v4f*>(out);

    const int stride = blockDim.x * gridDim.x;

    for (int i = blockIdx.x * blockDim.x + threadIdx.x; i < n; i += stride) {
        // Prefetch next iteration's coords (gfx1250 global_prefetch_b8).
        int inext = i + stride;
        if (inext < n) {
            __builtin_prefetch(&x[3 * inext], 0, 3);
        }

        // Streaming (last-use) coordinate loads.
        float cx = __builtin_nontemporal_load(&x[3 * i + 0]);
        float cy = __builtin_nontemporal_load(&x[3 * i + 1]);
        float cz = __builtin_nontemporal_load(&x[3 * i + 2]);

        // p = clip(x * NBINS, 0, NBINS-1)
        float px = fminf(fmaxf(cx * (float)NBINS, 0.0f), (float)NB1);
        float py = fminf(fmaxf(cy * (float)NBINS, 0.0f), (float)NB1);
        float pz = fminf(fmaxf(cz * (float)NBINS, 0.0f), (float)NB1);

        // lower corner (floor; p >= 0 so int cast == floor), upper clamped
        int ix0 = (int)px, iy0 = (int)py, iz0 = (int)pz;
        int ix1 = min(ix0 + 1, NB1);
        int iy1 = min(iy0 + 1, NB1);
        int iz1 = min(iz0 + 1, NB1);

        float fx = px - (float)ix0;
        float fy = py - (float)iy0;
        float fz = pz - (float)iz0;

        // Row bases in v4f units: ((ix*128 + iy) * 128)
        int r00 = ((ix0 << 7) + iy0) << 7;
        int r01 = ((ix0 << 7) + iy1) << 7;
        int r10 = ((ix1 << 7) + iy0) << 7;
        int r11 = ((ix1 << 7) + iy1) << 7;

        // Issue all 8 gathers (each an aligned global_load_b128) before use
        // so they form one load clause and overlap the weight VALU work.
        v4f c000 = g4[r00 + iz0];
        v4f c001 = g4[r00 + iz1];
        v4f c010 = g4[r01 + iz0];
        v4f c011 = g4[r01 + iz1];
        v4f c100 = g4[r10 + iz0];
        v4f c101 = g4[r10 + iz1];
        v4f c110 = g4[r11 + iz0];
        v4f c111 = g4[r11 + iz1];

        // Weights in the reference's order/associativity: (wx*wy)*wz.
        float wx0 = 1.0f - fx, wx1 = fx;
        float wy0 = 1.0f - fy, wy1 = fy;
        float wz0 = 1.0f - fz, wz1 = fz;

        v4f acc = ((wx0 * wy0) * wz0) * c000;
        acc += ((wx0 * wy0) * wz1) * c001;
        acc += ((wx0 * wy1) * wz0) * c010;
        acc += ((wx0 * wy1) * wz1) * c011;
        acc += ((wx1 * wy0) * wz0) * c100;
        acc += ((wx1 * wy0) * wz1) * c101;
        acc += ((wx1 * wy1) * wz0) * c110;
        acc += ((wx1 * wy1) * wz1) * c111;

        // Streaming store (th:NT) — output is written once, never re-read;
        // keep L2 capacity for the grid.
        __builtin_nontemporal_store(acc, &o4[i]);
    }
}

extern "C" void kernel_launch(void* const* d_in, const int* in_sizes, int n_in,
                              void* d_out, int out_size, void* d_ws, size_t ws_size,
                              hipStream_t stream) {
    const float* x    = (const float*)d_in[0];   // [N,3] fp32
    const float* grid = (const float*)d_in[1];   // [128,128,128,4] fp32
    float* out        = (float*)d_out;           // [N,4] fp32

    const int n = in_sizes[0] / 3;               // 2,000,000 points

    const int threads = 256;                     // 8 wave32 per workgroup
    int blocks = 4096;                           // ~1M threads; ~2 pts/thread
    int needed = (n + threads - 1) / threads;
    if (blocks > needed) blocks = needed;

    grid_enc3d_trilerp<<<blocks, threads, 0, stream>>>(x, grid, out, n);
}